// SE3Attention_80736795230622
// MI455X (gfx1250) — compile-verified
//
#include <hip/hip_runtime.h>
#include <math.h>

#define H    8
#define D    16
#define HD   128
#define NNB  32
#define NE   32
#define NJ   34           // null + self + 32 neighbors
#define SCALE0   0.25f                 // 1/sqrt(16)
#define SCALE1   0.14433756729740643f  // 1/sqrt(48)
#define SHARED_S 0.7071067811865476f   // sqrt(0.5)
#define NEGV    -1e9f

typedef __attribute__((ext_vector_type(2))) float v2f;
typedef __attribute__((ext_vector_type(8))) float v8f;
typedef __attribute__((__vector_size__(16))) int b128_t;   // 128-bit payload

#define AS1 __attribute__((address_space(1)))
#define AS3 __attribute__((address_space(3)))

#if defined(__has_builtin)
#if __has_builtin(__builtin_amdgcn_global_load_async_to_lds_b128) && \
    __has_builtin(__builtin_amdgcn_s_wait_asynccnt)
#define USE_ASYNC_LDS 1
#endif
#endif
#ifndef USE_ASYNC_LDS
#define USE_ASYNC_LDS 0
#endif

__device__ __forceinline__ v8f wmma_f32(v2f a, v2f b, v8f c) {
  // D = A(16x4,f32) * B(4x16,f32) + C(16x16,f32)
  return __builtin_amdgcn_wmma_f32_16x16x4_f32(false, a, false, b, (short)0, c,
                                               false, false);
}

// Stage a 16-row x 128-float tile (row-major, LDS row stride 132) from global
// into LDS. Async (ASYNCcnt) when the toolchain exposes the builtin.
__device__ __forceinline__ void stage_tile_16x128(const float* __restrict__ g,
                                                  float* lds, int tid) {
#if USE_ASYNC_LDS
  for (int chunk = tid; chunk < 512; chunk += 128) {   // 512 x 16B
    int r  = chunk >> 5;            // 32 chunks per row
    int c4 = (chunk & 31) << 2;     // float column
    __builtin_amdgcn_global_load_async_to_lds_b128(
        (AS1 b128_t*)(g + (size_t)r * 128 + c4),
        (AS3 b128_t*)(lds + r * 132 + c4), 0, 0);
  }
  __builtin_amdgcn_s_wait_asynccnt(0);
#else
  for (int idx = tid; idx < 16 * 128; idx += 128) {
    int r = idx >> 7, c = idx & 127;
    lds[r * 132 + c] = g[(size_t)r * 128 + c];
  }
#endif
}

// ---------------------------------------------------------------------------
// Stage 1: self key/value projections.
//   skv0[node][o]      = sum_c Wkv0[o][c] * feats0[node][c]      (o<256,c<128)
//   skv1[node][o*3+m]  = sum_c Wkv1[o][c] * feats1[node][c][m]   (o<256,c<64)
// One block = 16-node tile, 4 waves. WMMA f32 16x16x4 over K.
// ---------------------------------------------------------------------------
__global__ void __launch_bounds__(128) selfkv_kernel(
    const float* __restrict__ feats0, const float* __restrict__ feats1,
    const float* __restrict__ Wkv0,   const float* __restrict__ Wkv1,
    float* __restrict__ skv0_ws,      float* __restrict__ skv1_ws)
{
  __shared__ float A0[16 * 132];        // feats0 tile, padded stride
  __shared__ float A1[3 * 16 * 68];     // feats1 tile per m, padded stride
  const int tid = threadIdx.x;
  const int nb  = blockIdx.x * 16;

  stage_tile_16x128(feats0 + (size_t)nb * 128, A0, tid);
  for (int idx = tid; idx < 16 * 192; idx += 128) {
    int r = idx / 192, rem = idx - r * 192;   // rem = c*3 + m
    int c = rem / 3, m = rem - c * 3;
    A1[(m * 16 + r) * 68 + c] = feats1[(size_t)(nb + r) * 192 + rem];
  }
  __syncthreads();

  const int wave = tid >> 5, lane = tid & 31;
  const int r    = lane & 15;           // A row / B col / D col
  const int kk   = (lane >> 4) << 1;    // K sub-offset: 0 or 2
  const int mrow = (lane >> 4) << 3;    // D row base: 0 or 8

  // ---- deg-0: 16 output tiles of Wkv0, K = 128 (compile-time) ----
  const float* arow0 = A0 + r * 132;
  for (int t = wave; t < 16; t += 4) {
    const int oBase = t << 4;
    const float* brow = Wkv0 + (size_t)(oBase + r) * 128;
    v8f c = {0.f, 0.f, 0.f, 0.f, 0.f, 0.f, 0.f, 0.f};
#pragma unroll
    for (int k0 = 0; k0 < 128; k0 += 4)
      c = wmma_f32(*(const v2f*)(arow0 + k0 + kk),
                   *(const v2f*)(brow + k0 + kk), c);
#pragma unroll
    for (int v = 0; v < 8; ++v)
      skv0_ws[(size_t)(nb + mrow + v) * 256 + oBase + r] = c[v];
  }

  // ---- deg-1: 16 tiles x 3 m of Wkv1, K = 64 (compile-time) ----
  for (int q = wave; q < 48; q += 4) {
    const int m     = q >> 4;
    const int oBase = (q & 15) << 4;
    const float* arow = A1 + (m * 16 + r) * 68;
    const float* brow = Wkv1 + (size_t)(oBase + r) * 64;
    v8f c = {0.f, 0.f, 0.f, 0.f, 0.f, 0.f, 0.f, 0.f};
#pragma unroll
    for (int k0 = 0; k0 < 64; k0 += 4)
      c = wmma_f32(*(const v2f*)(arow + k0 + kk),
                   *(const v2f*)(brow + k0 + kk), c);
#pragma unroll
    for (int v = 0; v < 8; ++v)
      skv1_ws[(size_t)(nb + mrow + v) * 768 + (oBase + r) * 3 + m] = c[v];
  }
}

// ---------------------------------------------------------------------------
// Stage 2: attention core. One block per node (256 thr = 8 waves = 8 heads).
// Phase 1: lane j computes logit j (lanes 0,1 also j+32), shfl softmax.
// Phase 2: lane = output component, shuffle-broadcast attn weights.
// ---------------------------------------------------------------------------
__global__ void __launch_bounds__(256) attn_kernel(
    const float* __restrict__ q0, const float* __restrict__ k0,
    const float* __restrict__ v0, const float* __restrict__ q1,
    const float* __restrict__ k1, const float* __restrict__ v1,
    const float* __restrict__ edges, const float* __restrict__ W_bias,
    const float* __restrict__ self_bias, const float* __restrict__ null_bias,
    const float* __restrict__ null_k0, const float* __restrict__ null_v0,
    const float* __restrict__ null_k1, const float* __restrict__ null_v1,
    const float* __restrict__ head_w1, const unsigned char* __restrict__ nmask,
    const float* __restrict__ skv0_ws, const float* __restrict__ skv1_ws,
    float* __restrict__ out0_ws, float* __restrict__ out1_ws)
{
  __shared__ float q0_lds[H * D];
  __shared__ float q1_lds[H * 48];
  __shared__ float wb_lds[H * NE];

  const int i    = blockIdx.x;
  const int h    = threadIdx.x >> 5;
  const int lane = threadIdx.x & 31;

  // Stage per-head q and bias weights in LDS (same-wave producer/consumer).
  if (lane < D) q0_lds[h * D + lane] = q0[(size_t)i * HD + h * D + lane];
  for (int c = lane; c < 48; c += 32)
    q1_lds[h * 48 + c] = q1[((size_t)i * HD + h * D) * 3 + c];
  wb_lds[h * NE + lane] = W_bias[h * NE + lane];

  const float* q0l = &q0_lds[h * D];
  const float* q1l = &q1_lds[h * 48];
  const float* wbl = &wb_lds[h * NE];

  const float w1  = log1pf(expf(head_w1[h]));   // softplus
  const float sc1 = SCALE1 * w1;

  auto qk = [&](const float* kp0, const float* kp1) -> float {
    float s0 = 0.f, s1 = 0.f;
#pragma unroll
    for (int d = 0; d < D; d += 4) {
      float4 k4 = *reinterpret_cast<const float4*>(kp0 + d);
      s0 += q0l[d] * k4.x + q0l[d + 1] * k4.y + q0l[d + 2] * k4.z +
            q0l[d + 3] * k4.w;
    }
#pragma unroll
    for (int c = 0; c < 48; c += 4) {
      float4 k4 = *reinterpret_cast<const float4*>(kp1 + c);
      s1 += q1l[c] * k4.x + q1l[c + 1] * k4.y + q1l[c + 2] * k4.z +
            q1l[c + 3] * k4.w;
    }
    return s0 * SCALE0 + s1 * sc1;
  };
  auto ebias = [&](const float* ep) -> float {
    float eb = 0.f;
#pragma unroll
    for (int e = 0; e < NE; e += 4) {
      float4 e4 = *reinterpret_cast<const float4*>(ep + e);
      eb += wbl[e] * e4.x + wbl[e + 1] * e4.y + wbl[e + 2] * e4.z +
            wbl[e + 3] * e4.w;
    }
    return eb;
  };

  // ----- logits: slot 0 = j = lane -----
  float L0;
  {
    const float *kp0, *kp1;
    float bias;
    bool valid = true;
    if (lane == 0) {                       // null token
      kp0 = null_k0 + h * D;  kp1 = null_k1 + h * 48;  bias = null_bias[h];
    } else if (lane == 1) {                // self token
      kp0 = skv0_ws + (size_t)i * 256 + h * D;
      kp1 = skv1_ws + (size_t)i * 768 + h * 48;
      bias = self_bias[h];
    } else {                               // neighbor jn = lane-2
      size_t kb = (size_t)i * NNB + (lane - 2);
      kp0  = k0 + kb * HD + h * D;
      kp1  = k1 + (kb * HD + h * D) * 3;
      bias = ebias(edges + kb * NE);
      valid = nmask[kb] != 0;
    }
    float s = qk(kp0, kp1);
    L0 = valid ? (s + bias) * SHARED_S : NEGV;
  }
  // ----- slot 1 = j = lane + 32 (lanes 0,1 -> neighbors 30,31) -----
  float L1 = NEGV;
  if (lane < 2) {
    size_t kb = (size_t)i * NNB + (lane + 30);
    float s  = qk(k0 + kb * HD + h * D, k1 + (kb * HD + h * D) * 3);
    float eb = ebias(edges + kb * NE);
    L1 = (nmask[kb] != 0) ? (s + eb) * SHARED_S : NEGV;
  }

  // ----- wave softmax over 34 logits -----
  float mx = fmaxf(L0, L1);
  for (int s = 16; s > 0; s >>= 1) mx = fmaxf(mx, __shfl_xor(mx, s, 32));
  float e0 = expf(L0 - mx);
  float e1 = (lane < 2) ? expf(L1 - mx) : 0.f;
  float sm = e0 + e1;
  for (int s = 16; s > 0; s >>= 1) sm += __shfl_xor(sm, s, 32);
  const float inv = 1.f / sm;
  const float a0 = e0 * inv, a1 = e1 * inv;

  // ----- phase 2: weighted V sums -----
  // compA: lanes 0-15 -> deg0 d=lane ; lanes 16-31 -> deg1 c=lane-16
  // compB: all lanes  -> deg1 c=lane+16
  const int cA = lane - 16;
  const int cB = lane + 16;
  float accA = 0.f, accB = 0.f;
  for (int j = 0; j < NJ; ++j) {
    float w = (j < 32) ? __shfl(a0, j, 32) : __shfl(a1, j - 32, 32);
    const float *vp0, *vp1;
    if (j == 0) {
      vp0 = null_v0 + h * D;  vp1 = null_v1 + h * 48;
    } else if (j == 1) {
      vp0 = skv0_ws + (size_t)i * 256 + 128 + h * D;
      vp1 = skv1_ws + (size_t)i * 768 + 384 + h * 48;
    } else {
      size_t kb = (size_t)i * NNB + (j - 2);
      vp0 = v0 + kb * HD + h * D;
      vp1 = v1 + (kb * HD + h * D) * 3;
      if (j + 1 < NJ)   // stream hint for next neighbor's v rows
        __builtin_prefetch(v0 + (kb + 1) * HD + h * D, 0, 0);
    }
    float vA = (lane < 16) ? vp0[lane] : vp1[cA];
    float vB = vp1[cB];
    accA += w * vA;
    accB += w * vB;
  }
  if (lane < 16) out0_ws[(size_t)i * HD + h * D + lane] = accA;
  else           out1_ws[(size_t)i * 384 + h * 48 + cA] = accA;
  out1_ws[(size_t)i * 384 + h * 48 + cB] = accB;
}

// ---------------------------------------------------------------------------
// Stage 3: output projections with WMMA (K = 128 for both).
//   y0[node][o]    = sum_c Wo0[o][c] * out0[node][c]      (o,c < 128)
//   y1[node][o][m] = sum_c Wo1[o][c] * out1[node][c][m]   (o<64, c<128)
// ---------------------------------------------------------------------------
__global__ void __launch_bounds__(128) outproj_kernel(
    const float* __restrict__ out0_ws, const float* __restrict__ out1_ws,
    const float* __restrict__ Wo0,     const float* __restrict__ Wo1,
    float* __restrict__ y0,            float* __restrict__ y1)
{
  __shared__ float A0[16 * 132];
  __shared__ float A1[3 * 16 * 132];
  const int tid = threadIdx.x;
  const int nb  = blockIdx.x * 16;

  stage_tile_16x128(out0_ws + (size_t)nb * 128, A0, tid);
  for (int idx = tid; idx < 16 * 384; idx += 128) {
    int r = idx / 384, rem = idx - r * 384;   // rem = ch*3 + m
    int c = rem / 3, m = rem - c * 3;
    A1[(m * 16 + r) * 132 + c] = out1_ws[(size_t)(nb + r) * 384 + rem];
  }
  __syncthreads();

  const int wave = tid >> 5, lane = tid & 31;
  const int r    = lane & 15;
  const int kk   = (lane >> 4) << 1;
  const int mrow = (lane >> 4) << 3;

  for (int job = wave; job < 20; job += 4) {
    const float* arow;
    const float* brow;
    int oBase;
    if (job < 8) {                       // y0: 8 output tiles
      oBase = job << 4;
      arow  = A0 + r * 132;
      brow  = Wo0 + (size_t)(oBase + r) * 128;
    } else {                             // y1: 4 tiles x 3 m
      int q = job - 8;
      int m = q >> 2;
      oBase = (q & 3) << 4;
      arow  = A1 + (m * 16 + r) * 132;
      brow  = Wo1 + (size_t)(oBase + r) * 128;
    }
    v8f c = {0.f, 0.f, 0.f, 0.f, 0.f, 0.f, 0.f, 0.f};
#pragma unroll
    for (int k0 = 0; k0 < 128; k0 += 4)
      c = wmma_f32(*(const v2f*)(arow + k0 + kk),
                   *(const v2f*)(brow + k0 + kk), c);
    if (job < 8) {
#pragma unroll
      for (int v = 0; v < 8; ++v)
        y0[(size_t)(nb + mrow + v) * 128 + oBase + r] = c[v];
    } else {
      int m = (job - 8) >> 2;
#pragma unroll
      for (int v = 0; v < 8; ++v)
        y1[((size_t)(nb + mrow + v) * 64 + oBase + r) * 3 + m] = c[v];
    }
  }
}

// ---------------------------------------------------------------------------
extern "C" void kernel_launch(void* const* d_in, const int* in_sizes, int n_in,
                              void* d_out, int out_size, void* d_ws,
                              size_t ws_size, hipStream_t stream)
{
  const float* q0      = (const float*)d_in[0];
  const float* k0      = (const float*)d_in[1];
  const float* v0      = (const float*)d_in[2];
  const float* q1      = (const float*)d_in[3];
  const float* k1      = (const float*)d_in[4];
  const float* v1      = (const float*)d_in[5];
  const float* feats0  = (const float*)d_in[6];
  const float* feats1  = (const float*)d_in[7];
  const float* edges   = (const float*)d_in[8];
  const float* Wkv0    = (const float*)d_in[9];
  const float* Wkv1    = (const float*)d_in[10];
  const float* W_bias  = (const float*)d_in[11];
  const float* sbias   = (const float*)d_in[12];
  const float* nbias   = (const float*)d_in[13];
  const float* nk0     = (const float*)d_in[14];
  const float* nv0     = (const float*)d_in[15];
  const float* nk1     = (const float*)d_in[16];
  const float* nv1     = (const float*)d_in[17];
  const float* hw1     = (const float*)d_in[18];
  const float* Wo0     = (const float*)d_in[19];
  const float* Wo1     = (const float*)d_in[20];
  const unsigned char* nmask = (const unsigned char*)d_in[21];

  const int n = in_sizes[0] / HD;       // 2048

  float* skv0_ws = (float*)d_ws;                       // n*256
  float* skv1_ws = skv0_ws + (size_t)n * 256;          // n*768
  float* out0_ws = skv1_ws + (size_t)n * 768;          // n*128
  float* out1_ws = out0_ws + (size_t)n * 128;          // n*384
  float* y0 = (float*)d_out;                           // n*128
  float* y1 = y0 + (size_t)n * HD;                     // n*64*3

  selfkv_kernel<<<n / 16, 128, 0, stream>>>(feats0, feats1, Wkv0, Wkv1,
                                            skv0_ws, skv1_ws);
  attn_kernel<<<n, 256, 0, stream>>>(q0, k0, v0, q1, k1, v1, edges, W_bias,
                                     sbias, nbias, nk0, nv0, nk1, nv1, hw1,
                                     nmask, skv0_ws, skv1_ws, out0_ws, out1_ws);
  outproj_kernel<<<n / 16, 128, 0, stream>>>(out0_ws, out1_ws, Wo0, Wo1,
                                             y0, y1);
}